// InteractionAccelerationModule_5102421147906
// MI455X (gfx1250) — compile-verified
//
#include <hip/hip_runtime.h>
#include <hip/hip_bf16.h>
#include <math.h>

// ---------------------------------------------------------------------------
// InteractionAccelerationModule for MI455X (gfx1250)
//
// Bandwidth-bound: 128 MiB of partial_grads streamed once (~6us at 23.3TB/s;
// ~3us exploiting the triangular mask). The (j,c) contraction (K=32, fp32) is
// done with V_WMMA_F32_16X16X4_F32:
//   C[k_local, n] += A[k_local, jc] * B[jc, n],  B nonzero only in cols 0/1
// so C[:,0] accumulates the x-gradient and C[:,1] the y-gradient for 16 k's.
// Per-k triangular masking is folded into the A fragment per-lane (EXEC stays
// all-ones, as WMMA requires). Weight coefficients are precomputed so pg is
// read exactly once. coefC staging uses GLOBAL_LOAD_ASYNC_TO_LDS_B128
// (ASYNCcnt) and each lane prefetches its next sequential row segment.
// ---------------------------------------------------------------------------

#define T_DIM  1024
#define DT_    0.4f
#define EPS_   1e-6f
#define CT     128      // t'-chunk length per block
#define NCHUNK (T_DIM / CT)

typedef __attribute__((ext_vector_type(2))) float v2f;
typedef __attribute__((ext_vector_type(8))) float v8f;

// True LDS byte offset of a __shared__ object (addrspace(3) cast).
__device__ __forceinline__ unsigned lds_offset_of(const void* p) {
  return (unsigned)(unsigned long long)(__attribute__((address_space(3))) const void*)p;
}

// ---------------------------------------------------------------------------
// Kernel 1: ws[j,t,c] = ( sum_i 1/||diff[i,j,t,:]|| ) * ( sum_m diff[j,m,t,c] )
// diff = central 2nd difference (zero-padded ends) of ado minus ado_wo.
// One thread per (t, j).  Layout of ws_g: [t][j][c] -> t*32 + j*2 + c.
// ---------------------------------------------------------------------------
__global__ __launch_bounds__(256)
void k_ws(const float* __restrict__ ado, const float* __restrict__ ado_wo,
          float* __restrict__ ws_g) {
  int tid = blockIdx.x * 256 + threadIdx.x;   // T_DIM*16 threads
  int t = tid >> 4;
  int j = tid & 15;
  float w = 0.f, s0 = 0.f, s1 = 0.f;
  if (t > 0 && t < T_DIM - 1) {
    const float inv_dt2 = 1.0f / (DT_ * DT_);
    for (int m = 0; m < 16; ++m) {
      // s[j,t,c] accumulation: diff[a=j, mode=m, t, c]
      {
        int base = ((j * 16 + m) * T_DIM + t) * 2;
        float dx = ((ado[base - 2] - 2.f * ado[base]     + ado[base + 2])
                 -  (ado_wo[base - 2] - 2.f * ado_wo[base]     + ado_wo[base + 2])) * inv_dt2;
        float dy = ((ado[base - 1] - 2.f * ado[base + 1] + ado[base + 3])
                 -  (ado_wo[base - 1] - 2.f * ado_wo[base + 1] + ado_wo[base + 3])) * inv_dt2;
        s0 += dx; s1 += dy;
      }
      // w[j,t] accumulation: norm_inv[i=m, j, t]
      {
        int base = ((m * 16 + j) * T_DIM + t) * 2;
        float dx = ((ado[base - 2] - 2.f * ado[base]     + ado[base + 2])
                 -  (ado_wo[base - 2] - 2.f * ado_wo[base]     + ado_wo[base + 2])) * inv_dt2;
        float dy = ((ado[base - 1] - 2.f * ado[base + 1] + ado[base + 3])
                 -  (ado_wo[base - 1] - 2.f * ado_wo[base + 1] + ado_wo[base + 3])) * inv_dt2;
        float nn = sqrtf(dx * dx + dy * dy);
        w += (nn < EPS_) ? 0.f : (1.f / nn);
      }
    }
  }
  ws_g[t * 32 + j * 2 + 0] = w * s0;
  ws_g[t * 32 + j * 2 + 1] = w * s1;
}

// ---------------------------------------------------------------------------
// Kernel 2: interior coefficient  coefC[t][jc] = ws[t+1][jc] - 2*ws[t][jc]
// (valid for t <= T-3; zero elsewhere)
// ---------------------------------------------------------------------------
__global__ __launch_bounds__(256)
void k_coef(const float* __restrict__ ws_g, float* __restrict__ coef_g) {
  int i = blockIdx.x * 256 + threadIdx.x;     // T_DIM*32 threads
  int t = i >> 5;
  float v = 0.f;
  if (t <= T_DIM - 3) v = ws_g[i + 32] - 2.f * ws_g[i];
  coef_g[i] = v;
}

// ---------------------------------------------------------------------------
// Kernel 3: main streaming WMMA kernel.
// Grid: x = 64 k-tiles (16 k each), y = 8 t'-chunks (128 t' each).
// Block: 256 threads = 8 waves; wave w handles t' = cs + w + 8*ii.
// A fragment (16x4 f32): lane L holds row M=L%16, K = 2*(L/16)+v  -> this
// lane loads float2 at jc = 4p + 2h for WMMA step p (8 steps cover K=32).
// Partial sums written to scratch, reduced in fixed order by kernel 4.
// ---------------------------------------------------------------------------
__global__ __launch_bounds__(256)
void k_main(const float* __restrict__ pg, const float* __restrict__ ws_g,
            const float* __restrict__ coef_g, float* __restrict__ partials) {
  const int kb  = blockIdx.x * 16;        // k-tile base
  const int cs  = blockIdx.y * CT;        // chunk start t'
  const int tid = threadIdx.x;
  float* pout = partials + blockIdx.y * (T_DIM * 2) + kb * 2;  // 32 floats

  // Chunk fully below the triangular band -> contribute zeros.
  if (cs + CT - 1 < kb - 2) {
    if (tid < 32) pout[tid] = 0.f;
    return;
  }

  __shared__ __align__(16) float lds_coef[CT * 32];  // 16 KB staged coefC
  __shared__ float lds_red[8 * 32];                  // per-wave partials

  // Async memory->LDS staging of this chunk's coefC (ASYNCcnt path).
  {
    const float* src = coef_g + (size_t)cs * 32;
    const unsigned lds0 = lds_offset_of(&lds_coef[0]);
    for (int i = tid * 4; i < CT * 32; i += 256 * 4) {
      unsigned dst = lds0 + (unsigned)(i * 4);               // byte offset
      unsigned long long gaddr = (unsigned long long)(src + i);
      asm volatile("global_load_async_to_lds_b128 %0, %1, off"
                   :: "v"(dst), "v"(gaddr) : "memory");
    }
    asm volatile("s_wait_asynccnt 0x0" ::: "memory");
  }
  __syncthreads();

  const int wid  = tid >> 5;
  const int lane = tid & 31;
  const int h    = lane >> 4;             // K half (A/B fragment)
  const int n    = lane & 15;             // M row (A) / N column (B)
  const int row  = kb + n;                // this lane's k

  v8f acc = {};                           // C/D accumulator (16x16 f32)

  for (int ii = 0; ii < CT / 8; ++ii) {
    const int tp = cs + wid + 8 * ii;     // wave-uniform t'
    if (tp > T_DIM - 2) break;            // t'=T-1 contributes only via ws[.,0,.] == 0
    if (tp < kb - 2) continue;            // no lane in band yet

    // Load this lane's half-row of pg[row, tp, :, :] (32 contiguous floats).
    const float* rowp = pg + ((size_t)row * T_DIM + tp) * 32;
    // Keep this lane's sequential HBM stream deep (next t' is +8 -> +1KB).
    __builtin_prefetch(rowp + 8 * 32, 0, 0);
    v2f r[8];
#pragma unroll
    for (int p = 0; p < 8; ++p) r[p] = *(const v2f*)(rowp + 4 * p + 2 * h);

    // Pass 1: interior band t' in [k-1, T-3], coefficient ws[t'+1]-2ws[t'].
    if (tp <= T_DIM - 3 && tp >= kb - 1) {
      const float m1 = (row >= 1 && tp >= row - 1) ? 1.f : 0.f;
      const float* cbase = lds_coef + (tp - cs) * 32;
#pragma unroll
      for (int p = 0; p < 8; ++p) {
        v2f a; a.x = r[p].x * m1; a.y = r[p].y * m1;
        v2f cf = *(const v2f*)(cbase + 4 * p + 2 * h);
        v2f b; b.x = (n == 0) ? cf.x : 0.f;   // even jc -> c=0 -> column 0
               b.y = (n == 1) ? cf.y : 0.f;   // odd  jc -> c=1 -> column 1
        acc = __builtin_amdgcn_wmma_f32_16x16x4_f32(
            false, a, false, b, (short)0, acc, false, false);
      }
    }

    // Pass 2: band edges (wave-uniform branch; per-lane masks in A).
    const bool eb = (tp == T_DIM - 2);                  // coef = -2*ws[T-2]
    const bool e1 = (tp >= kb - 2) && (tp <= kb + 13);  // t' == k-2 for some lane
    if (eb || e1) {
      float m2, scale;
      int tw;
      if (eb) { m2 = (row >= 1 && tp >= row - 1) ? 1.f : 0.f; tw = T_DIM - 2; scale = -2.f; }
      else    { m2 = (tp == row - 2) ? 1.f : 0.f;             tw = tp + 1;    scale =  1.f; }
      const float* wbase = ws_g + (size_t)tw * 32;      // small, L2-resident
#pragma unroll
      for (int p = 0; p < 8; ++p) {
        v2f a; a.x = r[p].x * m2; a.y = r[p].y * m2;
        v2f cf = *(const v2f*)(wbase + 4 * p + 2 * h);
        v2f b; b.x = (n == 0) ? scale * cf.x : 0.f;
               b.y = (n == 1) ? scale * cf.y : 0.f;
        acc = __builtin_amdgcn_wmma_f32_16x16x4_f32(
            false, a, false, b, (short)0, acc, false, false);
      }
    }
  }

  // Extract columns 0/1 of C: lane (h, n<2), VGPR rr holds (M=rr+8h, N=n).
  if (n < 2) {
#pragma unroll
    for (int rr = 0; rr < 8; ++rr)
      lds_red[wid * 32 + (rr + 8 * h) * 2 + n] = acc[rr];
  }
  __syncthreads();
  if (tid < 32) {
    float sum = 0.f;
#pragma unroll
    for (int w = 0; w < 8; ++w) sum += lds_red[w * 32 + tid];
    pout[tid] = sum;   // partials[chunk][kb + tid/2][tid%2]
  }
}

// ---------------------------------------------------------------------------
// Kernel 4: fixed-order reduction over chunks (deterministic), k=0 -> 0.
// ---------------------------------------------------------------------------
__global__ __launch_bounds__(256)
void k_reduce(const float* __restrict__ partials, float* __restrict__ out) {
  int i = blockIdx.x * 256 + threadIdx.x;    // 2*T_DIM threads
  int k = i >> 1;
  float s = 0.f;
  if (k >= 1)
    for (int ch = 0; ch < NCHUNK; ++ch) s += partials[ch * (T_DIM * 2) + i];
  out[i] = s;
}

// ---------------------------------------------------------------------------
extern "C" void kernel_launch(void* const* d_in, const int* in_sizes, int n_in,
                              void* d_out, int out_size, void* d_ws, size_t ws_size,
                              hipStream_t stream) {
  (void)in_sizes; (void)n_in; (void)out_size; (void)ws_size;
  const float* ado    = (const float*)d_in[0];  // [16,16,1024,2]
  const float* ado_wo = (const float*)d_in[1];  // [16,16,1024,2]
  const float* pg     = (const float*)d_in[2];  // [1024,1024,16,2]
  float* out = (float*)d_out;                   // [2048]

  float* wsf    = (float*)d_ws;
  float* ws_g   = wsf;                 // 32768 floats: ws[t][j][c]
  float* coef_g = wsf + 32768;         // 32768 floats: coefC[t][jc]
  float* part_g = wsf + 65536;         // NCHUNK*2048 floats: partials

  k_ws    <<<dim3(T_DIM * 16 / 256), dim3(256), 0, stream>>>(ado, ado_wo, ws_g);
  k_coef  <<<dim3(T_DIM * 32 / 256), dim3(256), 0, stream>>>(ws_g, coef_g);
  k_main  <<<dim3(64, NCHUNK),       dim3(256), 0, stream>>>(pg, ws_g, coef_g, part_g);
  k_reduce<<<dim3(2 * T_DIM / 256),  dim3(256), 0, stream>>>(part_g, out);
}